// VMRNNCell_10840497455219
// MI455X (gfx1250) — compile-verified
//
#include <hip/hip_runtime.h>
#include <hip/hip_bf16.h>

// Problem constants (from reference)
#define B_  4
#define L_  2048
#define DM  256          // D_MODEL
#define DI  512          // D_INNER
#define DS  16           // D_STATE
#define DTR 16           // DT_RANK
#define HID_ 512

typedef __attribute__((ext_vector_type(2))) float v2f;
typedef __attribute__((ext_vector_type(8))) float v8f;
typedef __attribute__((ext_vector_type(4))) unsigned int v4u;
typedef __attribute__((ext_vector_type(8))) int v8i_;
typedef __attribute__((ext_vector_type(4))) int v4i;

__device__ __forceinline__ float sigmoidf_(float x) {
  return 1.0f / (1.0f + __expf(-x));
}

// ---------------------------------------------------------------------------
// Tensor Data Mover: DMA a 2D tile (rows x rowlen fp32, row stride = `stride`
// elements) from global memory into LDS. D# built per CDNA5 ISA:
//   group0: count=1 | lds_addr | global_addr[56:0] | type=2
//   group1: data_size=4B, tensor_dim0/1, tile_dim0/1, tensor_dim0_stride
// Groups 2/3 zero (2D tensor). Completion tracked with TENSORcnt.
// This toolchain exposes the 6-arg builtin:
//   (uint32x4, int32x8, int32x4, int32x4, int32x8, i32 cpol)
// ---------------------------------------------------------------------------
__device__ __forceinline__ void tdm_load_2d_to_lds(const float* gsrc, float* lds_dst,
                                                   unsigned rows, unsigned rowlen,
                                                   unsigned stride) {
  const unsigned long long ga = (unsigned long long)(const void*)gsrc;
  const unsigned lad = (unsigned)(unsigned long long)(void*)lds_dst; // LDS byte offset
  v4u g0;
  g0.x = 1u;                                                  // count=1, is_restore=0
  g0.y = lad;                                                 // lds_addr
  g0.z = (unsigned)(ga & 0xFFFFFFFFull);                      // global_addr[31:0]
  g0.w = (unsigned)((ga >> 32) & 0x01FFFFFFull) | (2u << 30); // addr[56:32] | type=2
  v8i_ g1;
  g1[0] = (int)(2u << 16);                                        // data_size = 4 bytes
  g1[1] = (int)((rowlen & 0xFFFFu) << 16);                        // tensor_dim0[15:0]
  g1[2] = (int)(((rowlen >> 16) & 0xFFFFu) | ((rows & 0xFFFFu) << 16)); // t_dim0[31:16] | t_dim1[15:0]
  g1[3] = (int)(((rows >> 16) & 0xFFFFu) | ((rowlen & 0xFFFFu) << 16)); // t_dim1[31:16] | tile_dim0
  g1[4] = (int)(rows & 0xFFFFu);                                  // tile_dim1 | tile_dim2=0
  g1[5] = (int)stride;                                            // tensor_dim0_stride[31:0]
  g1[6] = 0;                                                      // stride[47:32] | dim1_stride lo
  g1[7] = 0;
  const v4i  z4 = {0, 0, 0, 0};
  const v8i_ z8 = {0, 0, 0, 0, 0, 0, 0, 0};
  __builtin_amdgcn_tensor_load_to_lds(g0, g1, z4, z4, z8, 0);
}

// ---------------------------------------------------------------------------
// fp32 WMMA GEMM: each block computes a 128x16 output panel (8 waves, one
// 16x16 tile each along M, all sharing one 16-column B panel). The B panel
// (16 x K fp32) is staged into LDS once per block by the TDM.
// A row-major (lda); Bm is N x K row-major (B[k][n] = Bm[n*ldb + k]).
// EPI 0: plain store.  EPI 1: softplus(acc + bias[n]) (dt path).
// ---------------------------------------------------------------------------
template <int EPI, int K>
__global__ void gemm_wmma_kernel(const float* __restrict__ A, int lda,
                                 const float* __restrict__ Bm, int ldb,
                                 float* __restrict__ C, int ldc,
                                 int N, const float* __restrict__ bias) {
  __shared__ float sB[16 * K];                 // 16-row B panel, row-major
  const int lane = threadIdx.x & 31;
  const int wave = threadIdx.x >> 5;
  const int tilesN = N >> 4;
  const int tN = (blockIdx.x % tilesN) << 4;
  const int tM = (((blockIdx.x / tilesN) << 3) + wave) << 4;

  if (threadIdx.x < 32) {                      // wave 0 issues the TDM DMA
    tdm_load_2d_to_lds(Bm + (size_t)tN * ldb, sB, 16u, (unsigned)K, (unsigned)ldb);
    __builtin_amdgcn_s_wait_tensorcnt(0);
  }
  __syncthreads();

  v8f acc = {0.f, 0.f, 0.f, 0.f, 0.f, 0.f, 0.f, 0.f};
  const int half = lane >> 4;                  // 0 or 1
  const int l16  = lane & 15;
  const float* arow = A  + (size_t)(tM + l16) * lda;
  const float* brow = sB + l16 * K;            // B row for N = tN + l16
  for (int k0 = 0; k0 < K; k0 += 4) {
    const int ka = k0 + half * 2;              // lanes 16-31 take K+2/K+3
    __builtin_prefetch(arow + k0 + 32, 0, 1);  // global_prefetch_b8 (speculative)
    v2f a = { arow[ka], arow[ka + 1] };        // global_load_b64
    v2f b = { brow[ka], brow[ka + 1] };        // ds_load_b64
    acc = __builtin_amdgcn_wmma_f32_16x16x4_f32(
        /*neg_a=*/false, a, /*neg_b=*/false, b,
        /*c_mod=*/(short)0, acc, /*reuse_a=*/false, /*reuse_b=*/false);
  }

  const int n = tN + l16;
  const float bn = (EPI == 1) ? bias[n] : 0.f;
#pragma unroll
  for (int r = 0; r < 8; ++r) {
    const int m = tM + (half ? 8 : 0) + r;
    float v = acc[r];
    if (EPI == 1) {                            // softplus(v + b_dt[n])
      v += bn;
      v = (v > 20.f) ? v : log1pf(__expf(v));
    }
    C[(size_t)m * ldc + n] = v;
  }
}

// ---------------------------------------------------------------------------
// Causal depthwise conv (width 4) + bias + SiLU:  xs[b,l,d]
// ---------------------------------------------------------------------------
__global__ void conv_silu_kernel(const float* __restrict__ xi,
                                 const float* __restrict__ conv_w,
                                 const float* __restrict__ conv_b,
                                 float* __restrict__ xs) {
  const size_t idx = (size_t)blockIdx.x * blockDim.x + threadIdx.x;
  if (idx >= (size_t)B_ * L_ * DI) return;
  const int d = (int)(idx % DI);
  const int l = (int)((idx / DI) % L_);
  const size_t b = idx / ((size_t)DI * L_);
  float acc = conv_b[d];
#pragma unroll
  for (int j = 0; j < 4; ++j) {
    const int ls = l + j - 3;
    const float xv = (ls >= 0) ? xi[((size_t)b * L_ + ls) * DI + d] : 0.f;
    acc = fmaf(conv_w[d * 4 + j], xv, acc);
  }
  xs[idx] = acc * sigmoidf_(acc);              // SiLU
}

// ---------------------------------------------------------------------------
// Selective scan, barrier-free: one thread per (b,d) channel, 16-state
// recurrence in registers. B_t (16 values) is loaded by lanes 0-15 of each
// wave (b is wave-uniform: 32 consecutive d's share a wave) and broadcast
// with __shfl -> no LDS, no __syncthreads, 64 fully independent waves.
// Only the final y (t = L-1) is materialized.
// ---------------------------------------------------------------------------
__global__ void scan_kernel(const float* __restrict__ xs,
                            const float* __restrict__ proj,   // (B,L,48): [dt|Bc|Cc]
                            const float* __restrict__ dt,
                            const float* __restrict__ A_log,
                            const float* __restrict__ Dp,
                            float* __restrict__ y_last) {
  const int gid  = blockIdx.x * blockDim.x + threadIdx.x;  // 0 .. B*DI-1
  const int b    = gid / DI;
  const int d    = gid % DI;
  const int l16  = threadIdx.x & 15;

  float Arow[DS], h[DS];
#pragma unroll
  for (int s = 0; s < DS; ++s) {
    Arow[s] = -__expf(A_log[d * DS + s]);
    h[s] = 0.f;
  }

  const size_t base = (size_t)b * L_;
  const float* dtp = dt   + base * DI + d;
  const float* xsp = xs   + base * DI + d;
  const float* pjp = proj + base * 48 + DTR + l16;   // B_t[l16]

  for (int t = 0; t < L_; ++t) {
    const float dtv = dtp[(size_t)t * DI];
    const float dx  = dtv * xsp[(size_t)t * DI];
    const float bv  = pjp[(size_t)t * 48];
    __builtin_prefetch(dtp + (size_t)(t + 1) * DI, 0, 1);
    __builtin_prefetch(xsp + (size_t)(t + 1) * DI, 0, 1);
    __builtin_prefetch(pjp + (size_t)(t + 1) * 48, 0, 1);
#pragma unroll
    for (int s = 0; s < DS; ++s) {
      const float bs = __shfl(bv, s, 32);      // broadcast B_t[s] within wave
      h[s] = fmaf(__expf(dtv * Arow[s]), h[s], bs * dx);
    }
  }

  const float cv = proj[(base + (L_ - 1)) * 48 + DTR + DS + l16];
  float y = 0.f;
#pragma unroll
  for (int s = 0; s < DS; ++s) y = fmaf(h[s], __shfl(cv, s, 32), y);
  y = fmaf(xsp[(size_t)(L_ - 1) * DI], Dp[d], y);
  y_last[gid] = y;
}

// ---------------------------------------------------------------------------
// Final fused kernel: z-gate (only last token needs the z half of W_in),
// W_out projection, LSTM cell. One block per batch, 512 threads.
// ---------------------------------------------------------------------------
__global__ void final_kernel(const float* __restrict__ x,
                             const float* __restrict__ W_in,
                             const float* __restrict__ y_last,
                             const float* __restrict__ W_out,
                             const float* __restrict__ W_ih,
                             const float* __restrict__ W_hh,
                             const float* __restrict__ b_ih,
                             const float* __restrict__ b_hh,
                             const float* __restrict__ h0,
                             const float* __restrict__ c0,
                             float* __restrict__ out) {
  const int b = blockIdx.x;
  const int tid = threadIdx.x;               // 0..511
  __shared__ float sy[DI];
  __shared__ float sm[DM];

  const float* xrow = x + ((size_t)b * L_ + (L_ - 1)) * DM;
  {
    // z[d] = x_last . W_in[512+d], y2 = y_last * silu(z)
    float z = 0.f;
    const float* wrow = W_in + (size_t)(DI + tid) * DM;
    for (int k = 0; k < DM; ++k) z = fmaf(xrow[k], wrow[k], z);
    sy[tid] = y_last[b * DI + tid] * z * sigmoidf_(z);
  }
  __syncthreads();

  if (tid < DM) {
    float acc = 0.f;
    const float* wrow = W_out + (size_t)tid * DI;
    for (int k = 0; k < DI; ++k) acc = fmaf(sy[k], wrow[k], acc);
    sm[tid] = acc;
  }
  __syncthreads();

  // LSTM gates for hidden index j = tid
  const int j = tid;
  float gi = b_ih[j]            + b_hh[j];
  float gf = b_ih[HID_ + j]     + b_hh[HID_ + j];
  float gg = b_ih[2 * HID_ + j] + b_hh[2 * HID_ + j];
  float go = b_ih[3 * HID_ + j] + b_hh[3 * HID_ + j];
  for (int k = 0; k < DM; ++k) {
    const float mv = sm[k];
    gi = fmaf(mv, W_ih[(size_t)j * DM + k], gi);
    gf = fmaf(mv, W_ih[(size_t)(HID_ + j) * DM + k], gf);
    gg = fmaf(mv, W_ih[(size_t)(2 * HID_ + j) * DM + k], gg);
    go = fmaf(mv, W_ih[(size_t)(3 * HID_ + j) * DM + k], go);
  }
  const float* h0row = h0 + (size_t)b * HID_;
  for (int k = 0; k < HID_; ++k) {
    const float hv = h0row[k];
    gi = fmaf(hv, W_hh[(size_t)j * HID_ + k], gi);
    gf = fmaf(hv, W_hh[(size_t)(HID_ + j) * HID_ + k], gf);
    gg = fmaf(hv, W_hh[(size_t)(2 * HID_ + j) * HID_ + k], gg);
    go = fmaf(hv, W_hh[(size_t)(3 * HID_ + j) * HID_ + k], go);
  }
  const float cn = sigmoidf_(gf) * c0[(size_t)b * HID_ + j] +
                   sigmoidf_(gi) * tanhf(gg);
  const float hn = sigmoidf_(go) * tanhf(cn);
  out[(size_t)b * HID_ + j] = hn;                       // h_new
  out[(size_t)B_ * HID_ + (size_t)b * HID_ + j] = cn;   // c_new
}

// ---------------------------------------------------------------------------
extern "C" void kernel_launch(void* const* d_in, const int* in_sizes, int n_in,
                              void* d_out, int out_size, void* d_ws, size_t ws_size,
                              hipStream_t stream) {
  const float* x      = (const float*)d_in[0];
  const float* h0     = (const float*)d_in[1];
  const float* c0     = (const float*)d_in[2];
  const float* W_in   = (const float*)d_in[3];
  const float* conv_w = (const float*)d_in[4];
  const float* conv_b = (const float*)d_in[5];
  const float* W_xproj= (const float*)d_in[6];
  const float* W_dt   = (const float*)d_in[7];
  const float* b_dt   = (const float*)d_in[8];
  const float* A_log  = (const float*)d_in[9];
  const float* Dp     = (const float*)d_in[10];
  const float* W_out  = (const float*)d_in[11];
  const float* W_ih   = (const float*)d_in[12];
  const float* W_hh   = (const float*)d_in[13];
  const float* b_ih   = (const float*)d_in[14];
  const float* b_hh   = (const float*)d_in[15];
  float* out = (float*)d_out;

  // Workspace layout (floats)
  float* ws    = (float*)d_ws;
  float* xi    = ws;                       // B*L*DI = 4,194,304
  float* xs    = ws + 4194304;             // B*L*DI
  float* proj  = ws + 8388608;             // B*L*48 = 393,216
  float* dt    = ws + 8781824;             // B*L*DI
  float* ylast = ws + 12976128;            // B*DI = 2048

  const int M = B_ * L_;                   // 8192  (M/128 = 64 panels)

  // 1) xi = x @ W_in[0:512].T   (M=8192, N=512, K=256)  — f32 WMMA + TDM B-panel
  gemm_wmma_kernel<0, DM><<<(M / 128) * (DI / 16), 256, 0, stream>>>(
      x, DM, W_in, DM, xi, DI, DI, nullptr);

  // 2) causal depthwise conv + SiLU
  conv_silu_kernel<<<(B_ * L_ * DI) / 256, 256, 0, stream>>>(xi, conv_w, conv_b, xs);

  // 3) proj = xs @ W_xproj.T    (M=8192, N=48, K=512)   — f32 WMMA + TDM B-panel
  gemm_wmma_kernel<0, DI><<<(M / 128) * (48 / 16), 256, 0, stream>>>(
      xs, DI, W_xproj, DI, proj, 48, 48, nullptr);

  // 4) dt = softplus(proj[:, :16] @ W_dt.T + b_dt)  (M=8192, N=512, K=16) — f32 WMMA
  gemm_wmma_kernel<1, DTR><<<(M / 128) * (DI / 16), 256, 0, stream>>>(
      proj, 48, W_dt, DTR, dt, DI, DI, b_dt);

  // 5) selective scan (sequential over L, barrier-free waves over b×d)
  scan_kernel<<<(B_ * DI) / 256, 256, 0, stream>>>(xs, proj, dt, A_log, Dp, ylast);

  // 6) z-gate + W_out + LSTM cell
  final_kernel<<<B_, HID_, 0, stream>>>(x, W_in, ylast, W_out, W_ih, W_hh,
                                        b_ih, b_hh, h0, c0, out);
}